// GCNSingleLayer_9818295239342
// MI455X (gfx1250) — compile-verified
//
#include <hip/hip_runtime.h>
#include <hip/hip_bf16.h>

typedef __attribute__((ext_vector_type(2))) float v2f;
typedef __attribute__((ext_vector_type(8))) float v8f;

#define GCN_N_NODES 50000
#define GCN_D_IN    256
#define GCN_D_OUT   96
#define GCN_ROW_TILES (GCN_N_NODES / 16)   // 3125, exact
#define GCN_COL_TILES (GCN_D_OUT / 16)     // 6
#define GCN_KCHUNK  64                     // K rows of W staged per LDS phase
#define GCN_KPAIRS  (GCN_KCHUNK / 2)       // 32 K-pair rows in LDS
#define GCN_PAIR_STRIDE 224                // floats per K-pair row; 224 % 64 == 32
                                           //  -> half-waves hit disjoint LDS banks

// ---------------------------------------------------------------------------
// Kernel 1: out[n, j] = b[j]   (bias init; atomics accumulate on top)
// ---------------------------------------------------------------------------
__global__ __launch_bounds__(256) void gcn_init_out(float* __restrict__ out,
                                                    const float* __restrict__ b) {
    int idx = blockIdx.x * 256 + threadIdx.x;
    if (idx < GCN_N_NODES * GCN_D_OUT) {
        out[idx] = b[idx % GCN_D_OUT];
    }
}

// ---------------------------------------------------------------------------
// Kernel 2: h = x @ W  via V_WMMA_F32_16X16X4_F32
//   grid: 625 blocks x 160 threads (5 waves) -> exactly 3125 row-tile waves.
//   Each wave computes a 16x96 strip of h (6 x v8f accumulators).
//   W staged in LDS as K-pair-interleaved pairs: Wlds[kpair][col][2] so each
//   lane's B fragment is one contiguous ds_load_b64 (no VGPR repacking).
// ---------------------------------------------------------------------------
__global__ __launch_bounds__(160) void gcn_gemm(const float* __restrict__ x,
                                                const float* __restrict__ W,
                                                float* __restrict__ h) {
    __shared__ float Wlds[GCN_KPAIRS * GCN_PAIR_STRIDE];   // 28 KB

    const int wave = threadIdx.x >> 5;               // 0..4
    const int lane = threadIdx.x & 31;
    const int rowTile = blockIdx.x * 5 + wave;       // 0..3124 (always valid)

    const int m   = lane & 15;                       // A row this lane owns
    const int kb  = (lane >> 4) << 1;                // K sub-offset: 0 or 2
    const int col = lane & 15;                       // B/D column this lane owns

    const float* xrow = x + (size_t)(rowTile * 16 + m) * GCN_D_IN;

    v8f acc[GCN_COL_TILES];
#pragma unroll
    for (int t = 0; t < GCN_COL_TILES; ++t) acc[t] = (v8f){};

    for (int k0 = 0; k0 < GCN_D_IN; k0 += GCN_KCHUNK) {
        // Stage W[k0 .. k0+64) x [0..96) into LDS, pair-interleaved:
        //   Wlds[(k>>1)*224 + c*2 + (k&1)] = W[k0+k][c]
        for (int i = threadIdx.x; i < GCN_KCHUNK * GCN_D_OUT; i += 160) {
            int k = i / GCN_D_OUT;
            int c = i - k * GCN_D_OUT;
            Wlds[(k >> 1) * GCN_PAIR_STRIDE + c * 2 + (k & 1)] =
                W[(size_t)(k0 + k) * GCN_D_OUT + c];
        }
        __syncthreads();

#pragma unroll 4
        for (int kk = 0; kk < GCN_KCHUNK; kk += 4) {
            // A fragment: 16x4 fp32; lane holds (m, k0+kk+kb) and (m, k0+kk+kb+1)
            // 8-byte aligned (kk+kb even, row base 1KB aligned) -> global_load_b64
            v2f a = *(const v2f*)&xrow[k0 + kk + kb];
            const int krow = (kk + kb) >> 1;         // K-pair row in LDS
#pragma unroll
            for (int t = 0; t < GCN_COL_TILES; ++t) {
                // B fragment: contiguous pair -> single ds_load_b64
                v2f bf = *(const v2f*)&Wlds[krow * GCN_PAIR_STRIDE + (t * 16 + col) * 2];
                acc[t] = __builtin_amdgcn_wmma_f32_16x16x4_f32(
                    false, a, false, bf, (short)0, acc[t], false, false);
            }
        }
        __syncthreads();
    }

    // D layout: VGPR r holds row M = r + 8*(lane>=16), col = lane&15
    const int rowBase = rowTile * 16 + ((lane >> 4) << 3);
#pragma unroll
    for (int t = 0; t < GCN_COL_TILES; ++t) {
#pragma unroll
        for (int r = 0; r < 8; ++r) {
            h[(size_t)(rowBase + r) * GCN_D_OUT + t * 16 + col] = acc[t][r];
        }
    }
}

// ---------------------------------------------------------------------------
// Kernel 3: scatter-add.  One wave per edge; lane handles 3 features.
//   out[dst[e], f] += h[src[e], f]  via global_atomic_add_f32 (no return).
// ---------------------------------------------------------------------------
__global__ __launch_bounds__(256) void gcn_scatter(const int* __restrict__ src,
                                                   const int* __restrict__ dst,
                                                   const float* __restrict__ h,
                                                   float* __restrict__ out,
                                                   int nEdges) {
    int e    = (int)((blockIdx.x * 256 + threadIdx.x) >> 5);
    int lane = threadIdx.x & 31;
    if (e >= nEdges) return;

    int s = src[e];
    int d = dst[e];
    const float* hs = h + (size_t)s * GCN_D_OUT;
    float*       od = out + (size_t)d * GCN_D_OUT;

#pragma unroll
    for (int i = 0; i < 3; ++i) {
        int f = lane + i * 32;   // 32*3 = 96 features
        (void)__hip_atomic_fetch_add(&od[f], hs[f],
                                     __ATOMIC_RELAXED, __HIP_MEMORY_SCOPE_AGENT);
    }
}

// ---------------------------------------------------------------------------
extern "C" void kernel_launch(void* const* d_in, const int* in_sizes, int n_in,
                              void* d_out, int out_size, void* d_ws, size_t ws_size,
                              hipStream_t stream) {
    const float* x    = (const float*)d_in[0];                 // [50000, 256]
    const int*   edge = (const int*)  d_in[1];                 // [2, 800000] int32
    const float* W    = (const float*)d_in[2];                 // [256, 96]
    const float* b    = (const float*)d_in[3];                 // [96]
    float*       out  = (float*)d_out;                         // [50000, 96]
    float*       h    = (float*)d_ws;                          // scratch: 19.2 MB

    const int E = in_sizes[1] / 2;

    // 1) out = broadcast(b)
    {
        int total  = GCN_N_NODES * GCN_D_OUT;
        int blocks = (total + 255) / 256;
        gcn_init_out<<<blocks, 256, 0, stream>>>(out, b);
    }
    // 2) h = x @ W   (WMMA fp32)
    {
        gcn_gemm<<<GCN_ROW_TILES / 5, 160, 0, stream>>>(x, W, h);
    }
    // 3) out[dst] += h[src]
    {
        long long waves  = E;                       // one wave per edge
        long long blocks = (waves * 32 + 255) / 256;
        gcn_scatter<<<(unsigned)blocks, 256, 0, stream>>>(edge, edge + E, h, out, E);
    }
}